// RotatedRoIPool_23055384445773
// MI455X (gfx1250) — compile-verified
//
#include <hip/hip_runtime.h>
#include <math.h>

#define NROIS  512
#define CCH    256
#define CSPLIT 4                    // channel splits per ROI (occupancy)
#define CPB    (CCH / CSPLIT)       // 64 channels per block
#define FHH    256
#define FWW    256
#define OUTS   7
#define GRIDP  14
#define NGRID  (GRIDP * GRIDP)      // 196
#define PLANE  (FHH * FWW)          // 65536
#define INV_SCALE 16.0f

typedef float        f32x2 __attribute__((ext_vector_type(2), aligned(4)));
typedef unsigned int u32x4 __attribute__((ext_vector_type(4)));
typedef int          i32x4 __attribute__((ext_vector_type(4)));
typedef int          i32x8 __attribute__((ext_vector_type(8)));

__device__ __forceinline__ float bilin_pair(const float* __restrict__ pl,
                                            int o0, int o1,
                                            float a0, float b0,
                                            float a1, float b1) {
  // Two b64 loads fetch both x-adjacent corners of each row at once.
  f32x2 p0 = *(const f32x2*)(pl + o0);
  f32x2 p1 = *(const f32x2*)(pl + o1);
  return fmaf(a0, p0.x, fmaf(b0, p0.y, fmaf(a1, p1.x, b1 * p1.y)));
}

__global__ __launch_bounds__(256) void rroi_pool_kernel(
    const float* __restrict__ features,
    const float* __restrict__ rois,
    float* __restrict__ out) {
  const int tid   = threadIdx.x;
  const int roi   = blockIdx.x;
  const int cbase = blockIdx.y * CPB;

  __shared__ float sroi[8];
  __shared__ float sA0[NGRID], sB0[NGRID], sA1[NGRID], sB1[NGRID];
  __shared__ int   sOff0[NGRID], sOff1[NGRID];

  // ---- Stage this ROI's 6 floats into LDS via the Tensor Data Mover ----
#if defined(__has_builtin)
#if __has_builtin(__builtin_amdgcn_tensor_load_to_lds)
  if (tid < 32) {  // wave 0 only issues the TDM op
    unsigned long long ga = (unsigned long long)(uintptr_t)(rois + (size_t)roi * 6);
    unsigned int ldsOff = (unsigned int)(uintptr_t)&sroi[0];  // low 32 bits = LDS byte offset
    u32x4 g0;
    g0[0] = 1u;                                            // count = 1 valid descriptor
    g0[1] = ldsOff;                                        // lds_addr
    g0[2] = (unsigned int)(ga & 0xffffffffull);            // global_addr lo
    g0[3] = (unsigned int)((ga >> 32) & 0x01ffffffull) | 0x80000000u;  // addr hi | type=2
    i32x8 g1;
    g1[0] = 0x00020000;   // data_size = 4 bytes; no multicast / barrier / iterate / pad
    g1[1] = 6 << 16;      // tensor_dim0 = 6
    g1[2] = 1 << 16;      // tensor_dim1 = 1
    g1[3] = 6 << 16;      // tile_dim0 = 6
    g1[4] = 0;
    g1[5] = 6;            // tensor_dim0_stride = 6
    g1[6] = 0;
    g1[7] = 0;
    i32x4 gz = {0, 0, 0, 0};
#if __clang_major__ >= 23
    i32x8 gz8 = {0, 0, 0, 0, 0, 0, 0, 0};
    __builtin_amdgcn_tensor_load_to_lds(g0, g1, gz, gz, gz8, 0);
#else
    __builtin_amdgcn_tensor_load_to_lds(g0, g1, gz, gz, 0);
#endif
#if __has_builtin(__builtin_amdgcn_s_wait_tensorcnt)
    __builtin_amdgcn_s_wait_tensorcnt((short)0);
#endif
  }
#endif
#endif
  __syncthreads();
  // Hedge: TDM descriptor semantics are probe-level; refresh with plain loads
  // (idempotent, ordered after the tensorcnt wait + barrier above).
  if (tid < 6) sroi[tid] = rois[(size_t)roi * 6 + tid];
  __syncthreads();

  const float rix = sroi[0];
  const float x1 = sroi[1], y1 = sroi[2], x2 = sroi[3], y2 = sroi[4], th = sroi[5];

  // ---- Phase 2: per-grid-point sample descriptors (shared by all channels) ----
  if (tid < NGRID) {
    const int gy = tid / GRIDP;
    const int gx = tid - gy * GRIDP;
    const float ux = (2.0f * (float)gx + 1.0f) * (1.0f / (float)GRIDP) - 1.0f;
    const float uy = (2.0f * (float)gy + 1.0f) * (1.0f / (float)GRIDP) - 1.0f;

    const float cx = (x1 + x2) * 0.5f, cy = (y1 + y2) * 0.5f;
    const float w = x2 - x1, h = y2 - y1;
    const float ct = cosf(th), st = sinf(th);

    const float px = cx + INV_SCALE * ( ct * w * 0.5f * ux + st * h * 0.5f * uy);
    const float py = cy + INV_SCALE * (-st * w * 0.5f * ux + ct * h * 0.5f * uy);
    // fw == fh == 256 makes the grid-normalize/denormalize collapse to x - 0.5
    const float ix = px - 0.5f;
    const float iy = py - 0.5f;

    const float fx = floorf(ix), fy = floorf(iy);
    const float wx = ix - fx,   wy = iy - fy;
    const int x0 = (int)fx, y0 = (int)fy;
    const int x1i = x0 + 1, y1i = y0 + 1;

    const bool vx0 = (x0 >= 0) && (x0 < FWW);
    const bool vx1 = (x1i >= 0) && (x1i < FWW);
    const bool vy0 = (y0 >= 0) && (y0 < FHH);
    const bool vy1 = (y1i >= 0) && (y1i < FHH);

    const float w00 = (vy0 && vx0) ? (1.0f - wy) * (1.0f - wx) : 0.0f;
    const float w01 = (vy0 && vx1) ? (1.0f - wy) * wx          : 0.0f;
    const float w10 = (vy1 && vx0) ? wy * (1.0f - wx)          : 0.0f;
    const float w11 = (vy1 && vx1) ? wy * wx                   : 0.0f;

    // Pair base so the two x-corners sit inside one b64 load.
    const int xb = min(max(x0, 0), FWW - 2);
    const bool selA = (x0 <= xb);  // value(x0) lives in pair.x
    const bool selB = (x0 <  xb);  // value(x1) lives in pair.x (only when x0 < 0)

    sA0[tid] = (selA ? w00 : 0.0f) + (selB ? w01 : 0.0f);
    sB0[tid] = (selA ? 0.0f : w00) + (selB ? 0.0f : w01);
    sA1[tid] = (selA ? w10 : 0.0f) + (selB ? w11 : 0.0f);
    sB1[tid] = (selA ? 0.0f : w10) + (selB ? 0.0f : w11);

    const int y0c = min(max(y0, 0), FHH - 1);
    const int y1c = min(max(y1i, 0), FHH - 1);
    sOff0[tid] = y0c * FWW + xb;
    sOff1[tid] = y1c * FWW + xb;
  }
  __syncthreads();

  // ---- Phase 3: gather + bilinear + 2x2 max over this block's channel slice ----
  if (tid < 245) {                    // 5 channel-groups x 49 outputs
    const int o  = tid % 49;
    const int cg = tid / 49;          // 0..4
    const int oy = o / 7;
    const int ox = o - oy * 7;
    const int g00 = (2 * oy) * GRIDP + 2 * ox;
    const int g01 = g00 + 1;
    const int g10 = g00 + GRIDP;
    const int g11 = g00 + GRIDP + 1;

    // Descriptors held in registers across the whole channel loop.
    const float a0A = sA0[g00], b0A = sB0[g00], a1A = sA1[g00], b1A = sB1[g00];
    const float a0B = sA0[g01], b0B = sB0[g01], a1B = sA1[g01], b1B = sB1[g01];
    const float a0C = sA0[g10], b0C = sB0[g10], a1C = sA1[g10], b1C = sB1[g10];
    const float a0D = sA0[g11], b0D = sB0[g11], a1D = sA1[g11], b1D = sB1[g11];
    const int oA0 = sOff0[g00], oA1 = sOff1[g00];
    const int oB0 = sOff0[g01], oB1 = sOff1[g01];
    const int oC0 = sOff0[g10], oC1 = sOff1[g10];
    const int oD0 = sOff0[g11], oD1 = sOff1[g11];

    const int n = (int)rix;
    const float* __restrict__ base = features + ((size_t)n * CCH) * PLANE;
    float* __restrict__ op = out + ((size_t)roi * CCH) * (OUTS * OUTS) + o;

    for (int c = cbase + cg; c < cbase + CPB; c += 5) {
      const float* __restrict__ pl = base + (size_t)c * PLANE;
      const float s0 = bilin_pair(pl, oA0, oA1, a0A, b0A, a1A, b1A);
      const float s1 = bilin_pair(pl, oB0, oB1, a0B, b0B, a1B, b1B);
      const float s2 = bilin_pair(pl, oC0, oC1, a0C, b0C, a1C, b1C);
      const float s3 = bilin_pair(pl, oD0, oD1, a0D, b0D, a1D, b1D);
      // Output is write-once / never re-read: NT store keeps the 192MB L2
      // dedicated to the (L2-resident) 134MB feature tensor.
      __builtin_nontemporal_store(fmaxf(fmaxf(s0, s1), fmaxf(s2, s3)),
                                  op + (size_t)c * (OUTS * OUTS));
    }
  }
}

extern "C" void kernel_launch(void* const* d_in, const int* in_sizes, int n_in,
                              void* d_out, int out_size, void* d_ws, size_t ws_size,
                              hipStream_t stream) {
  const float* features = (const float*)d_in[0];  // (2,256,256,256) f32
  const float* rois     = (const float*)d_in[1];  // (512,6) f32
  float* out            = (float*)d_out;          // (512,256,7,7) f32
  (void)in_sizes; (void)n_in; (void)out_size; (void)d_ws; (void)ws_size;

  dim3 grid(NROIS, CSPLIT);
  rroi_pool_kernel<<<grid, 256, 0, stream>>>(features, rois, out);
}